// DiscriminatorModelGRU_52527450030745
// MI455X (gfx1250) — compile-verified
//
#include <hip/hip_runtime.h>
#include <hip/hip_bf16.h>
#include <math.h>

// ---- problem constants (match reference) ----
#define NROWS 32768        // B*T = 256*128
#define EDIM  512
#define ADIM  18
#define EA    530          // E + A
#define HDIM  128
#define H3    384          // 3*H
#define FCDIM 256

typedef float v2f __attribute__((ext_vector_type(2)));
typedef float v8f __attribute__((ext_vector_type(8)));

__device__ __forceinline__ float fast_sigmoid(float x) {
    return 1.0f / (1.0f + __expf(-x));
}
__device__ __forceinline__ float fast_tanh(float x) {
    float xc = fminf(fmaxf(x, -10.0f), 10.0f);
    float e  = __expf(2.0f * xc);
    return (e - 1.0f) / (e + 1.0f);
}

// -------------------------------------------------------------------------
// Kernel 1: gi_pred / gi_true = concat(enc, act) @ Wih^T + bih   (fp32 WMMA)
// grid (NROWS/16, 3), block 256 (8 waves). Wave w -> col tile blockIdx.y*8+w.
// Main K range [0,512): branchless aligned float2 (b64) loads.
// Tail [512,532): clamped addresses + cndmask zero-pad; action columns are
// shared between the pred and true paths (one A load feeds both WMMAs).
// -------------------------------------------------------------------------
__global__ __launch_bounds__(256)
void gi_gemm_kernel(const float* __restrict__ enc_r, const float* __restrict__ act,
                    const float* __restrict__ enc_t, const float* __restrict__ Wih,
                    const float* __restrict__ bih,
                    float* __restrict__ gi_pred, float* __restrict__ gi_true)
{
    const int lane  = threadIdx.x & 31;
    const int wave  = threadIdx.x >> 5;
    const int row0  = blockIdx.x * 16;
    const int col0  = (blockIdx.y * 8 + wave) * 16;   // 0..368
    const int rsel  = lane & 15;
    const int khalf = (lane < 16) ? 0 : 2;            // A/B K-half per ISA layout
    const int arow  = row0 + rsel;
    const int bcol  = col0 + rsel;

    // all row bases are 8-byte aligned: EDIM, EA (=530) and kk are even
    const float2* __restrict__ pP = (const float2*)(enc_r + (size_t)arow * EDIM);
    const float2* __restrict__ pT = (const float2*)(enc_t + (size_t)arow * EDIM);
    const float2* __restrict__ pB = (const float2*)(Wih   + (size_t)bcol * EA);

    v8f accP = {0.f,0.f,0.f,0.f,0.f,0.f,0.f,0.f};
    v8f accT = {0.f,0.f,0.f,0.f,0.f,0.f,0.f,0.f};

#pragma unroll 4
    for (int k = 0; k < EDIM; k += 4) {
        const int i2 = (k + khalf) >> 1;
        const float2 fa = pP[i2];
        const float2 ft = pT[i2];
        const float2 fb = pB[i2];
        v2f aP; aP.x = fa.x; aP.y = fa.y;
        v2f aT; aT.x = ft.x; aT.y = ft.y;
        v2f b;  b.x  = fb.x; b.y  = fb.y;
        accP = __builtin_amdgcn_wmma_f32_16x16x4_f32(false, aP, false, b,
                                                     (short)0, accP, false, false);
        accT = __builtin_amdgcn_wmma_f32_16x16x4_f32(false, aT, false, b,
                                                     (short)0, accT, false, false);
    }

    // tail: columns 512..531 (18 action cols + 2 pad). Branchless:
    // clamp address into range, zero via cndmask when c >= EA.
    const float* __restrict__ actrow = act + (size_t)arow * ADIM;
    const float* __restrict__ wrow   = Wih + (size_t)bcol * EA;
#pragma unroll
    for (int k = EDIM; k < EDIM + 20; k += 4) {
        const int c0 = k + khalf;
        const int c1 = c0 + 1;
        const float a0 = actrow[min(c0 - EDIM, ADIM - 1)];
        const float a1 = actrow[min(c1 - EDIM, ADIM - 1)];
        const float b0 = wrow[min(c0, EA - 1)];
        const float b1 = wrow[min(c1, EA - 1)];
        v2f a; a.x = (c0 < EA) ? a0 : 0.f; a.y = (c1 < EA) ? a1 : 0.f;
        v2f b; b.x = (c0 < EA) ? b0 : 0.f; b.y = (c1 < EA) ? b1 : 0.f;
        accP = __builtin_amdgcn_wmma_f32_16x16x4_f32(false, a, false, b,
                                                     (short)0, accP, false, false);
        accT = __builtin_amdgcn_wmma_f32_16x16x4_f32(false, a, false, b,
                                                     (short)0, accT, false, false);
    }

    const float bias = bih[col0 + rsel];
#pragma unroll
    for (int r = 0; r < 8; ++r) {
        const int row = row0 + r + ((lane < 16) ? 0 : 8);
        const int idx = row * H3 + col0 + rsel;
        gi_pred[idx] = accP[r] + bias;
        gi_true[idx] = accT[r] + bias;
    }
}

// -------------------------------------------------------------------------
// Kernel 2: sequential GRU scan over NROWS steps. 1 block, 384 threads.
// Whh row register-resident as 64 VGPR pairs, mat-vec via v_pk_fma_f32
// (packed fp32, 64 instrs instead of 128). h + gh in LDS. First barrier is
// an explicit split barrier: signal, then issue next step's gi prefetch
// loads while other waves arrive, then wait.
// -------------------------------------------------------------------------
__global__ __launch_bounds__(384)
void gru_scan_kernel(const float* __restrict__ gi_pred, const float* __restrict__ gi_true,
                     const float* __restrict__ Whh, const float* __restrict__ bhh,
                     const float* __restrict__ h0, float* __restrict__ h_preds)
{
    __shared__ float4 h4[HDIM / 4];
    __shared__ float  gh_sm[H3];
    float* h_sm = (float*)h4;

    const int j = threadIdx.x;                 // 0..383
    const bool gate_thread = (j < HDIM);       // uniform per wave (waves 0-3)

    // Whh row j as 64 packed float2 pairs (128 VGPRs)
    v2f w2[HDIM / 2];
    {
        const float2* __restrict__ wr = (const float2*)(Whh + (size_t)j * HDIM);
#pragma unroll
        for (int k = 0; k < HDIM / 2; ++k) {
            const float2 t = wr[k];
            w2[k].x = t.x; w2[k].y = t.y;
        }
    }
    const float bj = bhh[j];

    if (gate_thread) h_sm[j] = h0[j];

    // prologue: load step-0 gi
    float gpr = 0.f, gpz = 0.f, gpn = 0.f, gtr = 0.f, gtz = 0.f, gtn = 0.f;
    if (gate_thread) {
        gpr = gi_pred[j];  gpz = gi_pred[j + HDIM];  gpn = gi_pred[j + 2 * HDIM];
        gtr = gi_true[j];  gtz = gi_true[j + HDIM];  gtn = gi_true[j + 2 * HDIM];
    }
    __syncthreads();

    for (int n = 0; n < NROWS; ++n) {
        // gh[j] = bhh[j] + Whh[j,:] . h   via packed-fp32 FMA (2 MACs/instr)
        v2f acc01 = {0.f, 0.f};
        v2f acc23 = {0.f, 0.f};
#pragma unroll
        for (int k4 = 0; k4 < HDIM / 4; ++k4) {
            const float4 hv = h4[k4];
            v2f h01; h01.x = hv.x; h01.y = hv.y;
            v2f h23; h23.x = hv.z; h23.y = hv.w;
            asm("v_pk_fma_f32 %0, %1, %2, %0"
                : "+v"(acc01) : "v"(w2[2 * k4 + 0]), "v"(h01));
            asm("v_pk_fma_f32 %0, %1, %2, %0"
                : "+v"(acc23) : "v"(w2[2 * k4 + 1]), "v"(h23));
        }
        gh_sm[j] = bj + ((acc01.x + acc01.y) + (acc23.x + acc23.y));

        // split barrier: make our gh store visible, signal arrival, then issue
        // the next step's gi prefetch while the other waves arrive.
        asm volatile("s_wait_dscnt 0x0" ::: "memory");
        asm volatile("s_barrier_signal -1" ::: "memory");

        float ngpr = 0.f, ngpz = 0.f, ngpn = 0.f, ngtr = 0.f, ngtz = 0.f, ngtn = 0.f;
        if (gate_thread && (n + 1 < NROWS)) {
            const float* gp = gi_pred + (size_t)(n + 1) * H3;
            const float* gt = gi_true + (size_t)(n + 1) * H3;
            ngpr = gp[j];  ngpz = gp[j + HDIM];  ngpn = gp[j + 2 * HDIM];
            ngtr = gt[j];  ngtz = gt[j + HDIM];  ngtn = gt[j + 2 * HDIM];
        }
        asm volatile("s_barrier_wait -1" ::: "memory");

        if (gate_thread) {
            const float hcur = h_sm[j];        // stable: only thread j writes h_sm[j]
            const float ghr = gh_sm[j];
            const float ghz = gh_sm[j + HDIM];
            const float ghn = gh_sm[j + 2 * HDIM];
            // graded (pred) path
            const float rp = fast_sigmoid(gpr + ghr);
            const float zp = fast_sigmoid(gpz + ghz);
            const float np = fast_tanh(gpn + rp * ghn);
            const float hp = (1.0f - zp) * np + zp * hcur;
            // state-update (true) path
            const float rt = fast_sigmoid(gtr + ghr);
            const float zt = fast_sigmoid(gtz + ghz);
            const float nt = fast_tanh(gtn + rt * ghn);
            const float hn = (1.0f - zt) * nt + zt * hcur;
            h_preds[(size_t)n * HDIM + j] = hp;
            h_sm[j] = hn;
        }
        // second barrier: h update must be visible before next mat-vec
        asm volatile("s_wait_dscnt 0x0" ::: "memory");
        asm volatile("s_barrier_signal -1" ::: "memory");
        asm volatile("s_barrier_wait -1" ::: "memory");

        gpr = ngpr; gpz = ngpz; gpn = ngpn;
        gtr = ngtr; gtz = ngtz; gtn = ngtn;
    }
}

// -------------------------------------------------------------------------
// Kernel 3: fused MLP head. grid NROWS/16, block 512 (16 waves = all 16
// fc1 col-tiles of one 16-row stripe). fc1 via fp32 WMMA -> relu -> LDS,
// fc2 as a block-wide dot: 8 cols/lane + wave32 shfl_xor reduction.
// -------------------------------------------------------------------------
__global__ __launch_bounds__(512)
void head_kernel(const float* __restrict__ h_preds, const float* __restrict__ fc1_w,
                 const float* __restrict__ fc1_b, const float* __restrict__ fc2_w,
                 const float* __restrict__ fc2_b, float* __restrict__ out)
{
    __shared__ float hid[16 * FCDIM];   // 16 KB

    const int lane  = threadIdx.x & 31;
    const int wave  = threadIdx.x >> 5;         // 0..15
    const int row0  = blockIdx.x * 16;
    const int col0  = wave * 16;
    const int rsel  = lane & 15;
    const int khalf = (lane < 16) ? 0 : 2;

    const float2* __restrict__ pa = (const float2*)(h_preds + (size_t)(row0 + rsel) * HDIM);
    const float2* __restrict__ pb = (const float2*)(fc1_w + (size_t)(col0 + rsel) * HDIM);

    v8f acc = {0.f,0.f,0.f,0.f,0.f,0.f,0.f,0.f};
#pragma unroll
    for (int k = 0; k < HDIM; k += 4) {
        const int i2 = (k + khalf) >> 1;
        const float2 fa = pa[i2];
        const float2 fb = pb[i2];
        v2f a; a.x = fa.x; a.y = fa.y;
        v2f b; b.x = fb.x; b.y = fb.y;
        acc = __builtin_amdgcn_wmma_f32_16x16x4_f32(false, a, false, b,
                                                    (short)0, acc, false, false);
    }

    const float bias = fc1_b[col0 + rsel];
#pragma unroll
    for (int r = 0; r < 8; ++r) {
        const int rl = r + ((lane < 16) ? 0 : 8);
        hid[rl * FCDIM + col0 + rsel] = fmaxf(acc[r] + bias, 0.0f);
    }
    __syncthreads();

    // fc2: wave w owns row w; each lane takes 8 columns, wave32 reduction
    {
        const float* hrow = &hid[wave * FCDIM];
        const int cbase = lane * 8;
        float s = 0.f;
#pragma unroll
        for (int c = 0; c < 8; ++c)
            s = fmaf(hrow[cbase + c], fc2_w[cbase + c], s);
#pragma unroll
        for (int off = 16; off > 0; off >>= 1)
            s += __shfl_xor(s, off, 32);
        if (lane == 0)
            out[row0 + wave] = fast_sigmoid(s + fc2_b[0]);
    }
}

// -------------------------------------------------------------------------
extern "C" void kernel_launch(void* const* d_in, const int* in_sizes, int n_in,
                              void* d_out, int out_size, void* d_ws, size_t ws_size,
                              hipStream_t stream) {
    (void)in_sizes; (void)n_in; (void)out_size; (void)ws_size;
    const float* enc_r = (const float*)d_in[0];   // rand_encoding [N,512]
    const float* act   = (const float*)d_in[1];   // actions       [N,18]
    const float* enc_t = (const float*)d_in[2];   // true_encoding [N,512]
    const float* Wih   = (const float*)d_in[3];   // [384,530]
    const float* Whh   = (const float*)d_in[4];   // [384,128]
    const float* bih   = (const float*)d_in[5];   // [384]
    const float* bhh   = (const float*)d_in[6];   // [384]
    const float* h0    = (const float*)d_in[7];   // [128]
    const float* fc1_w = (const float*)d_in[8];   // [256,128]
    const float* fc1_b = (const float*)d_in[9];   // [256]
    const float* fc2_w = (const float*)d_in[10];  // [1,256]
    const float* fc2_b = (const float*)d_in[11];  // [1]
    float* out = (float*)d_out;

    // workspace layout (fp32): gi_pred | gi_true | h_preds  (~117 MB)
    float* gi_pred = (float*)d_ws;
    float* gi_true = gi_pred + (size_t)NROWS * H3;
    float* h_preds = gi_true + (size_t)NROWS * H3;

    gi_gemm_kernel<<<dim3(NROWS / 16, 3), 256, 0, stream>>>(
        enc_r, act, enc_t, Wih, bih, gi_pred, gi_true);
    gru_scan_kernel<<<1, 384, 0, stream>>>(
        gi_pred, gi_true, Whh, bhh, h0, h_preds);
    head_kernel<<<NROWS / 16, 512, 0, stream>>>(
        h_preds, fc1_w, fc1_b, fc2_w, fc2_b, out);
}